// GCN_69647189671985
// MI455X (gfx1250) — compile-verified
//
#include <hip/hip_runtime.h>
#include <hip/hip_bf16.h>

typedef __attribute__((ext_vector_type(2))) float v2f;
typedef __attribute__((ext_vector_type(8))) float v8f;

#define IN_FEATS 128
#define N_HIDDEN 128
#define N_CLASSES 16

// ---------------------------------------------------------------------------
// K1: degree accumulation (float atomics; deg buffers pre-zeroed)
// ---------------------------------------------------------------------------
__global__ __launch_bounds__(256) void degree_kernel(const int* __restrict__ src,
                                                     const int* __restrict__ dst,
                                                     float* __restrict__ deg_out,
                                                     float* __restrict__ deg_in, int E) {
  int i = blockIdx.x * blockDim.x + threadIdx.x;
  if (i < E) {
    atomicAdd(&deg_out[src[i]], 1.0f);
    atomicAdd(&deg_in[dst[i]], 1.0f);
  }
}

// ---------------------------------------------------------------------------
// K2: norms = rsqrt(max(deg,1))
// ---------------------------------------------------------------------------
__global__ __launch_bounds__(256) void norm_kernel(const float* __restrict__ deg_out,
                                                   const float* __restrict__ deg_in,
                                                   float* __restrict__ out_norm,
                                                   float* __restrict__ in_norm, int N) {
  int i = blockIdx.x * blockDim.x + threadIdx.x;
  if (i < N) {
    out_norm[i] = rsqrtf(fmaxf(deg_out[i], 1.0f));
    in_norm[i]  = rsqrtf(fmaxf(deg_in[i], 1.0f));
  }
}

// ---------------------------------------------------------------------------
// K3: layer-1 edge aggregation: agg1[dst] += features[src] * out_norm[src]
//     One wave per edge; each lane owns 4 consecutive floats (b128 gather).
// ---------------------------------------------------------------------------
__global__ __launch_bounds__(256) void aggregate128_kernel(const float* __restrict__ X,
                                                           const float* __restrict__ out_norm,
                                                           const int* __restrict__ src,
                                                           const int* __restrict__ dst,
                                                           float* __restrict__ agg, int E) {
  const int lane = threadIdx.x & 31;              // 32 lanes x 4 floats = 128 feats
  const int sub  = threadIdx.x >> 5;              // 8 edges per 256-thread block
  const long e   = (long)blockIdx.x * 8 + sub;
  if (e < E) {
    const int s = src[e], d = dst[e];
    const float onorm = out_norm[s];
    const float4 x = *(const float4*)(X + (size_t)s * IN_FEATS + lane * 4);
    float* dstp = agg + (size_t)d * IN_FEATS + lane * 4;
    atomicAdd(dstp + 0, x.x * onorm);
    atomicAdd(dstp + 1, x.y * onorm);
    atomicAdd(dstp + 2, x.z * onorm);
    atomicAdd(dstp + 3, x.w * onorm);
  }
}

// ---------------------------------------------------------------------------
// K4: h1 = relu((agg1 @ W1) * in_norm[:,None] + b1)   M x 128 @ 128 x 128
//     f32 WMMA: each wave computes a 16x16 tile; 8 waves/block cover N=128.
// ---------------------------------------------------------------------------
__global__ __launch_bounds__(256) void gemm1_kernel(const float* __restrict__ A,
                                                    const float* __restrict__ W,
                                                    const float* __restrict__ bias,
                                                    const float* __restrict__ in_norm,
                                                    float* __restrict__ H, int M) {
  const int lane  = threadIdx.x & 31;
  const int wave  = threadIdx.x >> 5;
  const int tm    = blockIdx.x;                    // M tile
  const int tn    = wave;                          // N tile 0..7
  const int lo    = lane & 15;
  const int hi    = lane >> 4;                     // 0 or 1
  const int khalf = hi * 2;

  int rowA = tm * 16 + lo;
  if (rowA >= M) rowA = M - 1;                     // clamp: keeps EXEC all-ones for WMMA
  const int colB = tn * 16 + lo;
  const float* __restrict__ Arow = A + (size_t)rowA * IN_FEATS;

  v8f acc = {};
#pragma unroll 4
  for (int k0 = 0; k0 < IN_FEATS; k0 += 4) {
    v2f a, b;
    a.x = Arow[k0 + khalf + 0];
    a.y = Arow[k0 + khalf + 1];
    b.x = W[(k0 + khalf + 0) * N_HIDDEN + colB];
    b.y = W[(k0 + khalf + 1) * N_HIDDEN + colB];
    acc = __builtin_amdgcn_wmma_f32_16x16x4_f32(false, a, false, b, (short)0, acc,
                                                false, false);
  }

  const float bc = bias[colB];
  if (tm * 16 + 16 <= M) {                         // wave-uniform: full tile (always here)
#pragma unroll
    for (int i = 0; i < 8; ++i) {
      const int row = tm * 16 + i + hi * 8;
      H[(size_t)row * N_HIDDEN + colB] = fmaxf(acc[i] * in_norm[row] + bc, 0.0f);
    }
  } else {                                         // tail tile (generic M)
#pragma unroll
    for (int i = 0; i < 8; ++i) {
      const int row = tm * 16 + i + hi * 8;
      if (row < M)
        H[(size_t)row * N_HIDDEN + colB] = fmaxf(acc[i] * in_norm[row] + bc, 0.0f);
    }
  }
}

// ---------------------------------------------------------------------------
// K5: p = (h1 * out_norm[:,None]) @ W2    M x 128 @ 128 x 16
//     One wave per 16x16 output tile (N=16 fits in one tile).
// ---------------------------------------------------------------------------
__global__ __launch_bounds__(256) void gemm2_kernel(const float* __restrict__ H1,
                                                    const float* __restrict__ W2,
                                                    const float* __restrict__ out_norm,
                                                    float* __restrict__ P, int M,
                                                    int Mtiles) {
  const int lane = threadIdx.x & 31;
  const int wave = threadIdx.x >> 5;
  const int tm   = blockIdx.x * 8 + wave;
  if (tm >= Mtiles) return;                        // wave-uniform guard
  const int lo    = lane & 15;
  const int hi    = lane >> 4;
  const int khalf = hi * 2;

  int rowA = tm * 16 + lo;
  if (rowA >= M) rowA = M - 1;
  const float onorm = out_norm[rowA];
  const float* __restrict__ Arow = H1 + (size_t)rowA * N_HIDDEN;

  v8f acc = {};
#pragma unroll 4
  for (int k0 = 0; k0 < N_HIDDEN; k0 += 4) {
    v2f a, b;
    a.x = Arow[k0 + khalf + 0] * onorm;
    a.y = Arow[k0 + khalf + 1] * onorm;
    b.x = W2[(k0 + khalf + 0) * N_CLASSES + lo];
    b.y = W2[(k0 + khalf + 1) * N_CLASSES + lo];
    acc = __builtin_amdgcn_wmma_f32_16x16x4_f32(false, a, false, b, (short)0, acc,
                                                false, false);
  }

  if (tm * 16 + 16 <= M) {                         // wave-uniform: full tile
#pragma unroll
    for (int i = 0; i < 8; ++i) {
      const int row = tm * 16 + i + hi * 8;
      P[(size_t)row * N_CLASSES + lo] = acc[i];
    }
  } else {
#pragma unroll
    for (int i = 0; i < 8; ++i) {
      const int row = tm * 16 + i + hi * 8;
      if (row < M) P[(size_t)row * N_CLASSES + lo] = acc[i];
    }
  }
}

// ---------------------------------------------------------------------------
// K6: layer-2 edge aggregation: agg2[dst] += p[src]  (16 f32 per edge)
// ---------------------------------------------------------------------------
__global__ __launch_bounds__(256) void aggregate16_kernel(const float* __restrict__ P,
                                                          const int* __restrict__ src,
                                                          const int* __restrict__ dst,
                                                          float* __restrict__ agg, int E) {
  const int f   = threadIdx.x & 15;
  const int sub = threadIdx.x >> 4;                // 0..15
  long e = (long)blockIdx.x * 16 + sub;
  if (e < E) {
    const int s = src[e], d = dst[e];
    atomicAdd(&agg[(size_t)d * N_CLASSES + f], P[(size_t)s * N_CLASSES + f]);
  }
}

// ---------------------------------------------------------------------------
// K7: in-place epilogue: h2 = agg2 * in_norm[:,None] + b2
// ---------------------------------------------------------------------------
__global__ __launch_bounds__(256) void epilogue2_kernel(float* __restrict__ agg2,
                                                        const float* __restrict__ in_norm,
                                                        const float* __restrict__ b2,
                                                        int total) {
  int i = blockIdx.x * blockDim.x + threadIdx.x;
  if (i < total) {
    const int row = i >> 4, col = i & 15;
    agg2[i] = agg2[i] * in_norm[row] + b2[col];
  }
}

// ---------------------------------------------------------------------------
// K8: SDDMM scores: out[e] = dot16(h2[psrc],h2[pdst]); out[E+e] = neg
// ---------------------------------------------------------------------------
__global__ __launch_bounds__(256) void score_kernel(const float* __restrict__ H,
                                                    const int* __restrict__ ps,
                                                    const int* __restrict__ pd,
                                                    const int* __restrict__ ns,
                                                    const int* __restrict__ nd,
                                                    float* __restrict__ out, int E) {
  int e = blockIdx.x * blockDim.x + threadIdx.x;
  if (e >= E) return;
  {
    const float4* a = (const float4*)(H + (size_t)ps[e] * N_CLASSES);
    const float4* b = (const float4*)(H + (size_t)pd[e] * N_CLASSES);
    float s = 0.f;
#pragma unroll
    for (int j = 0; j < 4; ++j) {
      float4 x = a[j], y = b[j];
      s += x.x * y.x + x.y * y.y + x.z * y.z + x.w * y.w;
    }
    out[e] = s;
  }
  {
    const float4* a = (const float4*)(H + (size_t)ns[e] * N_CLASSES);
    const float4* b = (const float4*)(H + (size_t)nd[e] * N_CLASSES);
    float s = 0.f;
#pragma unroll
    for (int j = 0; j < 4; ++j) {
      float4 x = a[j], y = b[j];
      s += x.x * y.x + x.y * y.y + x.z * y.z + x.w * y.w;
    }
    out[(size_t)E + e] = s;
  }
}

// ---------------------------------------------------------------------------
extern "C" void kernel_launch(void* const* d_in, const int* in_sizes, int n_in,
                              void* d_out, int out_size, void* d_ws, size_t ws_size,
                              hipStream_t stream) {
  const float* features = (const float*)d_in[0];
  const int*   pos_src  = (const int*)d_in[1];
  const int*   pos_dst  = (const int*)d_in[2];
  const int*   neg_src  = (const int*)d_in[3];
  const int*   neg_dst  = (const int*)d_in[4];
  const float* W1       = (const float*)d_in[5];
  const float* b1       = (const float*)d_in[6];
  const float* W2       = (const float*)d_in[7];
  const float* b2       = (const float*)d_in[8];
  float*       out      = (float*)d_out;

  const int N = in_sizes[0] / IN_FEATS;   // 50000
  const int E = in_sizes[1];              // 1600000

  // Workspace layout (floats). First segment must be re-zeroed each call.
  float* w        = (float*)d_ws;
  float* deg_out  = w;                                    // N
  float* deg_in   = deg_out + N;                          // N
  float* agg1     = deg_in + N;                           // N*128
  float* agg2     = agg1 + (size_t)N * IN_FEATS;          // N*16  (becomes h2)
  float* out_norm = agg2 + (size_t)N * N_CLASSES;         // N
  float* in_norm  = out_norm + N;                         // N
  float* h1       = in_norm + N;                          // N*128
  float* p        = h1 + (size_t)N * N_HIDDEN;            // N*16

  const size_t zero_bytes = (size_t)N * (2 + IN_FEATS + N_CLASSES) * sizeof(float);
  hipMemsetAsync(d_ws, 0, zero_bytes, stream);

  // degrees -> norms
  degree_kernel<<<(E + 255) / 256, 256, 0, stream>>>(pos_src, pos_dst, deg_out, deg_in, E);
  norm_kernel<<<(N + 255) / 256, 256, 0, stream>>>(deg_out, deg_in, out_norm, in_norm, N);

  // layer 1: aggregate (with source normalization) then WMMA GEMM + epilogue
  aggregate128_kernel<<<(E + 7) / 8, 256, 0, stream>>>(features, out_norm, pos_src,
                                                       pos_dst, agg1, E);
  const int mtiles = (N + 15) / 16;       // 3125
  gemm1_kernel<<<mtiles, 256, 0, stream>>>(agg1, W1, b1, in_norm, h1, N);

  // layer 2: WMMA GEMM (project first) then aggregate, then epilogue
  gemm2_kernel<<<(mtiles + 7) / 8, 256, 0, stream>>>(h1, W2, out_norm, p, N, mtiles);
  aggregate16_kernel<<<(E + 15) / 16, 256, 0, stream>>>(p, pos_src, pos_dst, agg2, E);
  epilogue2_kernel<<<((N * N_CLASSES) + 255) / 256, 256, 0, stream>>>(agg2, in_norm, b2,
                                                                      N * N_CLASSES);

  // SDDMM edge scores on pos + neg graphs
  score_kernel<<<(E + 255) / 256, 256, 0, stream>>>(agg2, pos_src, pos_dst, neg_src,
                                                    neg_dst, out, E);
}